// Model_44976897524446
// MI455X (gfx1250) — compile-verified
//
#include <hip/hip_runtime.h>
#include <hip/hip_bf16.h>
#include <math.h>

// Shapes (compile-time constants from the reference)
#define B_   256
#define R_   20
#define W_   100
#define D_   128
#define H_   64
#define DH_  (D_ * H_)   // 8192

typedef float v2f __attribute__((ext_vector_type(2)));
typedef float v8f __attribute__((ext_vector_type(8)));

__device__ __forceinline__ float wave_reduce_sum32(float v) {
  #pragma unroll
  for (int off = 16; off > 0; off >>= 1) v += __shfl_xor(v, off, 32);
  return v;
}
__device__ __forceinline__ float wave_reduce_max32(float v) {
  #pragma unroll
  for (int off = 16; off > 0; off >>= 1) v = fmaxf(v, __shfl_xor(v, off, 32));
  return v;
}

// ---------------------------------------------------------------------------
// Kernel 1: doc_embed — one wave (32 lanes) per document.
// Lane l owns D-elements [4l, 4l+3]. Two passes over the W=100 gathered rows
// (second pass hits L1/L2: table is L2-resident). Scores staged in LDS.
// ---------------------------------------------------------------------------
__global__ __launch_bounds__(256) void doc_embed_kernel(
    const int*  __restrict__ words, const float* __restrict__ emb,
    const float* __restrict__ w_self, const float* __restrict__ b_self,
    float* __restrict__ out, int n_docs) {
  __shared__ float sc[8][W_];
  const int wave = threadIdx.x >> 5;
  const int lane = threadIdx.x & 31;
  const int doc  = blockIdx.x * 8 + wave;
  if (doc >= n_docs) return;               // whole-wave exit; no cross-wave sync used

  const float4 ws4 = *(const float4*)(w_self + lane * 4);
  const float  bb  = b_self[0];
  const int*   wp  = words + (size_t)doc * W_;

  // pass 1: per-word attention scores
  for (int w = 0; w < W_; ++w) {
    const int idx = wp[w];
    const float4 row = *(const float4*)(emb + (size_t)idx * D_ + lane * 4);
    float p = row.x * ws4.x + row.y * ws4.y + row.z * ws4.z + row.w * ws4.w;
    p = wave_reduce_sum32(p);
    if (lane == 0) sc[wave][w] = tanhf(p + bb);
  }
  // softmax over the 100 word scores (wave-cooperative)
  float m = -3.0e38f;
  for (int w = lane; w < W_; w += 32) m = fmaxf(m, sc[wave][w]);
  m = wave_reduce_max32(m);
  float s = 0.f;
  for (int w = lane; w < W_; w += 32) {
    const float e = expf(sc[wave][w] - m);
    sc[wave][w] = e;
    s += e;
  }
  s = wave_reduce_sum32(s);
  const float inv = 1.0f / s;

  // pass 2: weighted sum of rows (rows now cached)
  float4 acc = make_float4(0.f, 0.f, 0.f, 0.f);
  for (int w = 0; w < W_; ++w) {
    const int idx = wp[w];
    const float4 row = *(const float4*)(emb + (size_t)idx * D_ + lane * 4);
    const float g = sc[wave][w] * inv;
    acc.x += g * row.x; acc.y += g * row.y; acc.z += g * row.z; acc.w += g * row.w;
  }
  *(float4*)(out + (size_t)doc * D_ + lane * 4) = acc;
}

// ---------------------------------------------------------------------------
// Kernel 2: pq = tanh(q_emb(256x128) @ Wq(128x8192) + bq)  via WMMA f32 16x16x4
// One wave per 16x16 output tile; 8 waves/block.  K-loop: 32 steps of K=4.
// ---------------------------------------------------------------------------
__global__ __launch_bounds__(256) void pq_gemm_kernel(
    const float* __restrict__ Q, const float* __restrict__ Wq,
    const float* __restrict__ bq, float* __restrict__ pq) {
  const int wave = threadIdx.x >> 5;
  const int lane = threadIdx.x & 31;
  const int Mbase = blockIdx.y * 16;
  const int Nbase = (blockIdx.x * 8 + wave) * 16;
  const int lh = lane & 15;      // M (for A) / N (for B,C) within tile
  const int kq = lane >> 4;      // K half select (lanes 16-31 hold K+2,K+3)

  v8f c = {};
  const float* qrow = Q + (size_t)(Mbase + lh) * D_;
  #pragma unroll 4
  for (int k = 0; k < D_; k += 4) {
    const int ka = k + 2 * kq;
    v2f A; A.x = qrow[ka]; A.y = qrow[ka + 1];
    v2f Bv;
    Bv.x = Wq[(size_t)ka * DH_ + Nbase + lh];
    Bv.y = Wq[(size_t)(ka + 1) * DH_ + Nbase + lh];
    c = __builtin_amdgcn_wmma_f32_16x16x4_f32(false, A, false, Bv,
                                              (short)0, c, false, false);
  }
  const int n = Nbase + lh;
  const int mrow = Mbase + (lane < 16 ? 0 : 8);
  const float bqv = bq[n];
  #pragma unroll
  for (int i = 0; i < 8; ++i)
    pq[(size_t)(mrow + i) * DH_ + n] = tanhf(c[i] + bqv);
}

// ---------------------------------------------------------------------------
// Kernel 3: per-(batch, side) attention over reviews.
//   s(20x64) = rev(20x128) @ pq_b(128x64)  -> WMMA, M padded to 32 (2x4 tiles)
//   score[r] = s[r,:] . w_red  (16-lane shuffle + LDS ds_add_f32)
//   softmax over R=20, weighted pooling, fuse personalized = u + pf*q_emb.
// ---------------------------------------------------------------------------
__global__ __launch_bounds__(256) void attn_kernel(
    const float* __restrict__ rev_u, const float* __restrict__ rev_i,
    const float* __restrict__ pq,    const float* __restrict__ w_red,
    const float* __restrict__ qemb,  const float* __restrict__ pf_p,
    float* __restrict__ out) {
  __shared__ float score[32];
  __shared__ float wgt[32];
  const int b    = blockIdx.x;
  const int side = blockIdx.y;
  const float* rev = (side ? rev_i : rev_u) + (size_t)b * (R_ * D_);
  const float* pqb = pq + (size_t)b * DH_;
  const int tid  = threadIdx.x;
  const int wave = tid >> 5;
  const int lane = tid & 31;

  if (tid < 32) score[tid] = 0.f;
  __syncthreads();

  // 8 waves -> 2 M-tiles x 4 N-tiles
  const int mbase = (wave >> 2) * 16;
  const int hbase = (wave & 3) * 16;
  const int lh = lane & 15;
  const int kq = lane >> 4;
  const int m  = mbase + lh;       // review row this lane loads for A
  const bool mv = (m < R_);

  v8f c = {};
  #pragma unroll 4
  for (int k = 0; k < D_; k += 4) {
    const int ka = k + 2 * kq;
    v2f A;
    A.x = mv ? rev[m * D_ + ka]     : 0.f;
    A.y = mv ? rev[m * D_ + ka + 1] : 0.f;
    v2f Bv;
    Bv.x = pqb[ka * H_ + hbase + lh];         // pq[b, d=ka,   h]
    Bv.y = pqb[(ka + 1) * H_ + hbase + lh];   // pq[b, d=ka+1, h]
    c = __builtin_amdgcn_wmma_f32_16x16x4_f32(false, A, false, Bv,
                                              (short)0, c, false, false);
  }

  // reduce s[m, h] * w_red[h] over h within each 16-lane group
  const float wr = w_red[hbase + lh];
  #pragma unroll
  for (int i = 0; i < 8; ++i) {
    float v = c[i] * wr;
    #pragma unroll
    for (int off = 8; off > 0; off >>= 1) v += __shfl_xor(v, off, 32);
    const int mm = mbase + i + (lane >= 16 ? 8 : 0);
    if ((lane == 0 || lane == 16) && mm < R_) atomicAdd(&score[mm], v);
  }
  __syncthreads();

  // softmax over the 20 review scores (wave 0)
  if (wave == 0) {
    const float v = (lane < R_) ? score[lane] : -3.0e38f;
    float mx = v;
    #pragma unroll
    for (int off = 16; off > 0; off >>= 1) mx = fmaxf(mx, __shfl_xor(mx, off, 32));
    float e = (lane < R_) ? expf(v - mx) : 0.f;
    float s = e;
    #pragma unroll
    for (int off = 16; off > 0; off >>= 1) s += __shfl_xor(s, off, 32);
    wgt[lane] = e / s;
  }
  __syncthreads();

  // weighted pooling over reviews; fuse personalized term for user side
  if (tid < D_) {
    const int d = tid;
    float acc = 0.f;
    #pragma unroll
    for (int r = 0; r < R_; ++r) acc += wgt[r] * rev[r * D_ + d];
    if (side == 0) {
      const float pf = pf_p[0];
      out[(size_t)b * D_ + d] = acc + pf * qemb[(size_t)b * D_ + d];
    } else {
      out[(size_t)B_ * D_ + (size_t)b * D_ + d] = acc;
    }
  }
}

// ---------------------------------------------------------------------------
extern "C" void kernel_launch(void* const* d_in, const int* in_sizes, int n_in,
                              void* d_out, int out_size, void* d_ws, size_t ws_size,
                              hipStream_t stream) {
  (void)in_sizes; (void)n_in; (void)out_size; (void)ws_size;
  const int*   user   = (const int*)  d_in[0];  // (B,R,W) int32
  const int*   item   = (const int*)  d_in[1];  // (B,R,W) int32
  const int*   query  = (const int*)  d_in[2];  // (B,W)   int32
  const float* emb    = (const float*)d_in[3];  // (VOCAB,D)
  const float* w_self = (const float*)d_in[4];  // (D,1)
  const float* b_self = (const float*)d_in[5];  // (1,)
  const float* Wq     = (const float*)d_in[6];  // (D, D*H)
  const float* bq     = (const float*)d_in[7];  // (D*H,)
  const float* w_red  = (const float*)d_in[8];  // (H,1)
  const float* pf     = (const float*)d_in[9];  // scalar

  float* ws = (float*)d_ws;
  float* user_emb = ws;                                  // 256*20*128 = 655360
  float* item_emb = ws + (size_t)B_ * R_ * D_;           // 655360
  float* q_emb    = item_emb + (size_t)B_ * R_ * D_;     // 256*128 = 32768
  float* pq       = q_emb + (size_t)B_ * D_;             // 256*8192 = 2097152

  const int n_user = B_ * R_;  // 5120 docs
  const int n_item = B_ * R_;  // 5120 docs
  const int n_q    = B_;       // 256 docs  (all divisible by 8 waves/block)

  doc_embed_kernel<<<n_user / 8, 256, 0, stream>>>(user,  emb, w_self, b_self, user_emb, n_user);
  doc_embed_kernel<<<n_item / 8, 256, 0, stream>>>(item,  emb, w_self, b_self, item_emb, n_item);
  doc_embed_kernel<<<n_q    / 8, 256, 0, stream>>>(query, emb, w_self, b_self, q_emb,    n_q);

  // pq GEMM: M=256 (16 tiles), N=8192 (512 tiles); 8 N-tiles per block
  pq_gemm_kernel<<<dim3(DH_ / 16 / 8, B_ / 16), 256, 0, stream>>>(q_emb, Wq, bq, pq);

  // attention: one block per (batch, user/item)
  attn_kernel<<<dim3(B_, 2), 256, 0, stream>>>(user_emb, item_emb, pq, w_red,
                                               q_emb, pf, (float*)d_out);
}